// Encoder_8976481649194
// MI455X (gfx1250) — compile-verified
//
#include <hip/hip_runtime.h>

typedef __attribute__((ext_vector_type(2))) float v2f;
typedef __attribute__((ext_vector_type(8))) float v8f;

#define IN_DIM 128
#define HID 128
#define K2 (2 * IN_DIM)     // 256
#define WSTRIDE (K2 + 4)    // 260: pad so lane-halves split across all 64 LDS banks

// ---------------------------------------------------------------------------
// 1) zero workspace (norm/deg + agg) -- must be re-zeroed every launch
// ---------------------------------------------------------------------------
__global__ void zero_ws_kernel(float* __restrict__ p, int n) {
    int i = blockIdx.x * blockDim.x + threadIdx.x;
    int stride = gridDim.x * blockDim.x;
    for (; i < n; i += stride) p[i] = 0.0f;
}

// ---------------------------------------------------------------------------
// 2) in-degree via f32 atomics (exact for counts << 2^24)
// ---------------------------------------------------------------------------
__global__ void deg_kernel(const int* __restrict__ dst, float* __restrict__ deg, int E) {
    int e = blockIdx.x * blockDim.x + threadIdx.x;
    if (e < E) atomicAdd(&deg[dst[e]], 1.0f);
}

// ---------------------------------------------------------------------------
// 3) norm[i] = clamp(deg,1)^-0.5  (in place on the deg buffer)
// ---------------------------------------------------------------------------
__global__ void norm_kernel(float* __restrict__ deg_norm, int n) {
    int i = blockIdx.x * blockDim.x + threadIdx.x;
    if (i < n) {
        float d = fmaxf(deg_norm[i], 1.0f);
        deg_norm[i] = 1.0f / sqrtf(d);
    }
}

// ---------------------------------------------------------------------------
// 4) edge scatter: agg[dst] += h[src] * norm[src]
//    one wave32 per edge, float4 per lane (32*16B = 512B row), f32 atomics.
//    h (51MB) and agg (51MB) both fit in the 192MB L2 -> L2-bw bound.
// ---------------------------------------------------------------------------
__global__ void scatter_kernel(const float* __restrict__ h, const float* __restrict__ norm,
                               const int* __restrict__ src, const int* __restrict__ dst,
                               float* __restrict__ agg, int E) {
    int e = blockIdx.x * (blockDim.x >> 5) + (threadIdx.x >> 5);
    int lane = threadIdx.x & 31;
    if (e >= E) return;
    int s = src[e];
    int d = dst[e];
    float ns = norm[s];
    float4 v = ((const float4*)(h + (size_t)s * IN_DIM))[lane];
    float* ap = agg + (size_t)d * IN_DIM + lane * 4;
    atomicAdd(ap + 0, v.x * ns);
    atomicAdd(ap + 1, v.y * ns);
    atomicAdd(ap + 2, v.z * ns);
    atomicAdd(ap + 3, v.w * ns);
}

// ---------------------------------------------------------------------------
// 5) fused GEMM + bias + row-L2-normalize using fp32 WMMA (16x16x4).
//    - W staged TRANSPOSED in LDS: sWt[n][k], row stride 260 floats, so a
//      lane's B pair (W[kk][n], W[kk+1][n]) is one aligned ds_load_b64.
//    - K loop split in two straight-line halves (h part, agg*norm part) so
//      there is no branch inside the WMMA loop.
//    - each wave computes a full 16x128 output block in 8 v8f accumulators;
//      bias + sum-of-squares + L2 normalize all stay in registers.
// ---------------------------------------------------------------------------
__global__ void __launch_bounds__(256)
tag_gemm_kernel(const float* __restrict__ h, const float* __restrict__ agg,
                const float* __restrict__ norm, const float* __restrict__ weight,
                const float* __restrict__ bias, float* __restrict__ out, int n) {
    __shared__ float sWt[HID * WSTRIDE];   // [n=0..127][k=0..255 (+4 pad)]
    __shared__ float sBias[HID];

    const int tid = threadIdx.x;
    // coalesced read of weight[k][n], transposed write to sWt[n][k]
    for (int idx = tid; idx < K2 * HID; idx += 256) {
        int k  = idx >> 7;      // / 128
        int nn = idx & 127;
        sWt[nn * WSTRIDE + k] = weight[idx];
    }
    if (tid < HID) sBias[tid] = bias[tid];
    __syncthreads();

    const int wave = tid >> 5;
    const int lane = tid & 31;
    const int half = lane >> 4;   // 0: K+0/K+1 (A) & rows 0-7 (C/D); 1: K+2/K+3 & rows 8-15
    const int lr   = lane & 15;   // A row / B-C-D column within tile
    const int m0   = (blockIdx.x * 8 + wave) * 16;
    if (m0 >= n) return;

    const int row = m0 + lr;
    const float nrm = norm[row];
    const float* hrow = h   + (size_t)row * IN_DIM;
    const float* arow = agg + (size_t)row * IN_DIM;
    const float* sWl  = sWt + lr;          // this lane's column family base

    v8f acc[8] = {};

    // ---- K in [0,128): A from h --------------------------------------------
#pragma unroll 2
    for (int k0 = 0; k0 < IN_DIM; k0 += 4) {
        const int kk = k0 + 2 * half;                 // lane's K pair: kk, kk+1
        v2f a = *(const v2f*)(hrow + kk);
#pragma unroll
        for (int t = 0; t < 8; ++t) {
            v2f b = *(const v2f*)(sWl + t * 16 * WSTRIDE + kk);   // ds_load_b64
            acc[t] = __builtin_amdgcn_wmma_f32_16x16x4_f32(
                false, a, false, b, (short)0, acc[t], false, false);
        }
    }

    // ---- K in [128,256): A from agg * norm[row] ----------------------------
#pragma unroll 2
    for (int k0 = 0; k0 < IN_DIM; k0 += 4) {
        const int kk = k0 + 2 * half;
        v2f t2 = *(const v2f*)(arow + kk);
        v2f a;
        a.x = t2.x * nrm;
        a.y = t2.y * nrm;
        const int kw = IN_DIM + kk;                   // B's K index
#pragma unroll
        for (int t = 0; t < 8; ++t) {
            v2f b = *(const v2f*)(sWl + t * 16 * WSTRIDE + kw);
            acc[t] = __builtin_amdgcn_wmma_f32_16x16x4_f32(
                false, a, false, b, (short)0, acc[t], false, false);
        }
    }

    // bias add + per-row sum of squares (row index within block = v + 8*half)
    float ss[8];
#pragma unroll
    for (int v = 0; v < 8; ++v) ss[v] = 0.0f;
#pragma unroll
    for (int t = 0; t < 8; ++t) {
        const float bv = sBias[t * 16 + lr];
#pragma unroll
        for (int v = 0; v < 8; ++v) {
            float x = acc[t][v] + bv;
            acc[t][v] = x;
            ss[v] += x * x;
        }
    }
    // reduce across the 16 lanes of each half (xor masks < 16 stay in-group)
#pragma unroll
    for (int off = 1; off < 16; off <<= 1) {
#pragma unroll
        for (int v = 0; v < 8; ++v) ss[v] += __shfl_xor(ss[v], off, 32);
    }
    float inv[8];
#pragma unroll
    for (int v = 0; v < 8; ++v) inv[v] = 1.0f / sqrtf(ss[v]);

#pragma unroll
    for (int t = 0; t < 8; ++t) {
#pragma unroll
        for (int v = 0; v < 8; ++v) {
            out[(size_t)(m0 + v + 8 * half) * HID + t * 16 + lr] = acc[t][v] * inv[v];
        }
    }
}

// ---------------------------------------------------------------------------
extern "C" void kernel_launch(void* const* d_in, const int* in_sizes, int n_in,
                              void* d_out, int out_size, void* d_ws, size_t ws_size,
                              hipStream_t stream) {
    const float* h      = (const float*)d_in[0];
    const float* weight = (const float*)d_in[1];
    const float* bias   = (const float*)d_in[2];
    const int*   src    = (const int*)d_in[3];
    const int*   dst    = (const int*)d_in[4];
    float*       out    = (float*)d_out;

    const int n = in_sizes[0] / IN_DIM;   // 100000 nodes
    const int E = in_sizes[3];            // 1600000 edges

    // ws layout: [0, n) = deg->norm, [n, n + n*128) = agg
    float* nrm = (float*)d_ws;
    float* agg = nrm + n;

    const int totalZero = n + n * IN_DIM;
    zero_ws_kernel<<<1024, 256, 0, stream>>>(nrm, totalZero);
    deg_kernel<<<(E + 255) / 256, 256, 0, stream>>>(dst, nrm, E);
    norm_kernel<<<(n + 255) / 256, 256, 0, stream>>>(nrm, n);
    // one wave32 per edge -> 8 edges per 256-thread block
    scatter_kernel<<<(E + 7) / 8, 256, 0, stream>>>(h, nrm, src, dst, agg, E);

    const int rowBlocks = (n + 15) / 16;          // 6250
    const int grid = (rowBlocks + 7) / 8;         // 8 waves per block
    tag_gemm_kernel<<<grid, 256, 0, stream>>>(h, agg, nrm, weight, bias, out, n);
}